// Damping_27066883900008
// MI455X (gfx1250) — compile-verified
//
#include <hip/hip_runtime.h>

typedef __attribute__((ext_vector_type(2))) float v2f;
typedef __attribute__((ext_vector_type(8))) float v8f;

#if __has_builtin(__builtin_amdgcn_tanhf)
#define HW_TANH(v) __builtin_amdgcn_tanhf(v)
#else
#define HW_TANH(v) tanhf(v)
#endif

// One wave (32 lanes) processes tiles of 16 batch rows.
// Activations live in WMMA C/D layout: VGPR v, lane l -> feature m = v + 8*(l>>4),
// batch column n = l & 15.  Two consecutive D VGPRs form a legal B operand for the
// next V_WMMA_F32_16X16X4_F32 (K rows paired (k, k+2) across lane halves), provided
// the weight A operand uses the matching feature permutation {2j,2j+1,2j+8,2j+9}.
__global__ __launch_bounds__(256) void Damping_27066883900008_kernel(
    const float* __restrict__ x,
    const float* __restrict__ w_d1, const float* __restrict__ w_d2,
    const float* __restrict__ w_d3, const float* __restrict__ w_o1,
    const float* __restrict__ w_o2, const float* __restrict__ w_o3,
    const float* __restrict__ b_d1, const float* __restrict__ b_d2,
    const float* __restrict__ b_d3, const float* __restrict__ b_o1,
    const float* __restrict__ b_o2, const float* __restrict__ b_o3,
    float* __restrict__ out, int nrows)
{
    const int lane = threadIdx.x & 31;   // wave32
    const int m    = lane & 15;          // row of A / batch column id
    const int h    = lane >> 4;          // lane half: selects K2/K3 (A) or feature+8

    // ---------------- per-wave constant setup (registers only) ----------------
    // Layer-1 A operand: K0,K1 = W1 rows 0,1 (lower half); upper half zeroed so
    // the zero B rows can never produce NaN*0.
    v2f a_d1, a_o1;
    a_d1.x = h ? 0.f : w_d1[0 * 16 + m];
    a_d1.y = h ? 0.f : w_d1[1 * 16 + m];
    a_o1.x = h ? 0.f : w_o1[0 * 16 + m];
    a_o1.y = h ? 0.f : w_o1[1 * 16 + m];

    // Layer-2 A operands, chunk j covers features {2j, 2j+1, 2j+8, 2j+9}.
    v2f a_d2[4], a_o2[4];
#pragma unroll
    for (int j = 0; j < 4; ++j) {
        a_d2[j].x = w_d2[(2 * j     + 8 * h) * 16 + m];
        a_d2[j].y = w_d2[(2 * j + 1 + 8 * h) * 16 + m];
        a_o2[j].x = w_o2[(2 * j     + 8 * h) * 16 + m];
        a_o2[j].y = w_o2[(2 * j + 1 + 8 * h) * 16 + m];
    }

    // Bias vectors pre-arranged in accumulator (C/D) layout: c[v] = bias[v + 8h].
    v8f cb_d1, cb_d2, cb_o1, cb_o2;
#pragma unroll
    for (int v = 0; v < 8; ++v) {
        cb_d1[v] = b_d1[v + 8 * h];
        cb_d2[v] = b_d2[v + 8 * h];
        cb_o1[v] = b_o1[v + 8 * h];
        cb_o2[v] = b_o2[v + 8 * h];
    }

    // Layer-3 weights, per lane: rows v + 8h (this lane's resident features).
    float w3a[8], w3b[8], w3c[8];
#pragma unroll
    for (int v = 0; v < 8; ++v) {
        w3a[v] = w_d3[(v + 8 * h) * 2 + 0];
        w3b[v] = w_d3[(v + 8 * h) * 2 + 1];
        w3c[v] = w_o3[(v + 8 * h)];
    }
    const float bd3_0 = b_d3[0], bd3_1 = b_d3[1], bo3_0 = b_o3[0];

    // ---------------- grid-stride over 16-row tiles ----------------
    const int ntiles = nrows >> 4;
    const int wave   = (int)((blockIdx.x * blockDim.x + threadIdx.x) >> 5);
    const int nwaves = (int)((gridDim.x * blockDim.x) >> 5);

    for (int t = wave; t < ntiles; t += nwaves) {
        const int row = (t << 4) + m;                 // both halves read same 16 rows
        const v2f xv  = *(const v2f*)(x + 2 * (size_t)row);

        // B operand for layer 1: K0=x0, K1=x1 in lower half, K2=K3=0 above.
        v2f bx;
        bx.x = h ? 0.f : xv.x;
        bx.y = h ? 0.f : xv.y;

        // ---- damping path ----
        v8f d = __builtin_amdgcn_wmma_f32_16x16x4_f32(false, a_d1, false, bx,
                                                      (short)0, cb_d1, false, false);
#pragma unroll
        for (int v = 0; v < 8; ++v) d[v] = HW_TANH(d[v]);

        v8f d2 = cb_d2;
#pragma unroll
        for (int j = 0; j < 4; ++j) {
            v2f bv; bv.x = d[2 * j]; bv.y = d[2 * j + 1];   // D VGPR pair == B operand
            d2 = __builtin_amdgcn_wmma_f32_16x16x4_f32(false, a_d2[j], false, bv,
                                                       (short)0, d2, false, false);
        }
#pragma unroll
        for (int v = 0; v < 8; ++v) d2[v] = HW_TANH(d2[v]);

        // ---- off-diagonal path ----
        v8f o = __builtin_amdgcn_wmma_f32_16x16x4_f32(false, a_o1, false, bx,
                                                      (short)0, cb_o1, false, false);
#pragma unroll
        for (int v = 0; v < 8; ++v) o[v] = HW_TANH(o[v]);

        v8f o2 = cb_o2;
#pragma unroll
        for (int j = 0; j < 4; ++j) {
            v2f bv; bv.x = o[2 * j]; bv.y = o[2 * j + 1];
            o2 = __builtin_amdgcn_wmma_f32_16x16x4_f32(false, a_o2[j], false, bv,
                                                       (short)0, o2, false, false);
        }
#pragma unroll
        for (int v = 0; v < 8; ++v) o2[v] = HW_TANH(o2[v]);

        // ---- layer 3 (16 -> 2 and 16 -> 1): VALU dot + cross-half reduce ----
        float pa = 0.f, pb = 0.f, pc = 0.f;
#pragma unroll
        for (int v = 0; v < 8; ++v) {
            pa = fmaf(w3a[v], d2[v], pa);
            pb = fmaf(w3b[v], d2[v], pb);
            pc = fmaf(w3c[v], o2[v], pc);
        }
        pa += __shfl_xor(pa, 16, 32);
        pb += __shfl_xor(pb, 16, 32);
        pc += __shfl_xor(pc, 16, 32);

        const float a3 = pa + bd3_0;
        const float b3 = pb + bd3_1;
        const float c  = pc + bo3_0;

        const float a = (fmaxf(a3, 0.f) + 0.001f) * xv.x;   // d3r[:,0]
        const float b = (fmaxf(b3, 0.f) + 0.001f) * xv.y;   // d3r[:,1]

        const float D0 = a * a * xv.x + a * c * xv.y;
        const float D1 = a * c * xv.x + (c * c + b * b) * xv.y;

        if (h == 0) {
            v2f ov; ov.x = D0; ov.y = D1;
            *(v2f*)(out + 2 * (size_t)row) = ov;
        }
    }
}

extern "C" void kernel_launch(void* const* d_in, const int* in_sizes, int n_in,
                              void* d_out, int out_size, void* d_ws, size_t ws_size,
                              hipStream_t stream) {
    const float* x    = (const float*)d_in[0];
    const float* w_d1 = (const float*)d_in[1];
    const float* w_d2 = (const float*)d_in[2];
    const float* w_d3 = (const float*)d_in[3];
    const float* w_o1 = (const float*)d_in[4];
    const float* w_o2 = (const float*)d_in[5];
    const float* w_o3 = (const float*)d_in[6];
    const float* b_d1 = (const float*)d_in[7];
    const float* b_d2 = (const float*)d_in[8];
    const float* b_d3 = (const float*)d_in[9];
    const float* b_o1 = (const float*)d_in[10];
    const float* b_o2 = (const float*)d_in[11];
    const float* b_o3 = (const float*)d_in[12];
    float* out = (float*)d_out;

    const int nrows  = in_sizes[0] / 2;       // x is (B, 2) flat
    const int ntiles = nrows / 16;

    // 256 threads = 8 waves/block; grid-stride so per-wave weight setup amortizes.
    int blocks = 4096;
    const int max_blocks = (ntiles + 7) / 8;  // >= 1 tile per wave
    if (blocks > max_blocks) blocks = max_blocks;
    if (blocks < 1) blocks = 1;

    Damping_27066883900008_kernel<<<blocks, 256, 0, stream>>>(
        x, w_d1, w_d2, w_d3, w_o1, w_o2, w_o3,
        b_d1, b_d2, b_d3, b_o1, b_o2, b_o3, out, nrows);
}